// ActionThenNodePolicy_57123065037301
// MI455X (gfx1250) — compile-verified
//
#include <hip/hip_runtime.h>
#include <hip/hip_bf16.h>
#include <math.h>

typedef __attribute__((ext_vector_type(2))) float v2f;
typedef __attribute__((ext_vector_type(8))) float v8f;

#define N_GRAPHS 2048
#define NPG 64
#define DD 256
#define AA 32
#define NEGV (-1000000000.0f)
#define EPSV 1e-20f

__device__ __forceinline__ float waveRedMax(float v) {
#pragma unroll
  for (int m = 16; m >= 1; m >>= 1) v = fmaxf(v, __shfl_xor(v, m, 32));
  return v;
}
__device__ __forceinline__ float waveRedSum(float v) {
#pragma unroll
  for (int m = 16; m >= 1; m >>= 1) v += __shfl_xor(v, m, 32);
  return v;
}

__global__ __launch_bounds__(256, 2)
void atnp_policy_kernel(const int* __restrict__ act,        // (2048,2)
                        const float* __restrict__ values,   // (131072,256)
                        const int* __restrict__ atm,        // (131072,32) bool-as-int
                        const int* __restrict__ aam,        // (131072,32) bool-as-int
                        const float* __restrict__ Wnode,    // (1,256)
                        const float* __restrict__ Wagn,     // (32,256)
                        const float* __restrict__ Wnga,     // (32,256)
                        const float* __restrict__ Wq,       // (64,256)
                        float* __restrict__ out_logprob,    // (2048)
                        float* __restrict__ out_entropy,    // (2048)
                        float* __restrict__ out_value,      // (2048)
                        float* __restrict__ out_pa,         // (2048,32)
                        float* __restrict__ out_pna)        // (131072,32)
{
  __shared__ float Lg[NPG][128];       // cols 0-31 agn, 32-63 nga, 64-127 q-raw (32 KB)
  __shared__ float nlPart[NPG][4];
  __shared__ float nodeLogit[NPG];
  __shared__ float p_n[NPG];
  __shared__ float red[8][AA];
  __shared__ float red2[8][AA];
  __shared__ float p_a[AA];
  __shared__ float colmax[AA], colsum[AA], sqp[AA], hsum[AA];

  const int g    = blockIdx.x;
  const int tid  = threadIdx.x;
  const int lane = tid & 31;
  const int w    = tid >> 5;           // 8 waves

  // ---------------- GEMM: Lg = V(64x256) @ W_all^T(256x128), fp32 WMMA -------------
  const int mtile   = w & 3;           // 4 M tiles of 16 nodes
  const int ng      = w >> 2;          // 2 N groups of 4 tiles
  const int halfSel = lane >> 4;       // 0 -> K pair {0,1}, 1 -> K pair {2,3}
  const int sub     = lane & 15;
  const int kSub    = halfSel * 2;

  // A fragment base: lane holds row (mtile*16+sub), K = k0 + kSub .. +1
  const float* aPtr = values + (size_t)(g * NPG + mtile * 16 + sub) * DD + kSub;

  // B fragment bases: lane holds W row (= output column) c, K = k0 + kSub .. +1
  const float* bPtr[4];
#pragma unroll
  for (int t = 0; t < 4; ++t) {
    int c = (ng * 4 + t) * 16 + sub;   // 0..127, tile-aligned so no boundary split
    const float* base; int r;
    if (c < 32)      { base = Wagn; r = c;      }
    else if (c < 64) { base = Wnga; r = c - 32; }
    else             { base = Wq;   r = c - 64; }
    bPtr[t] = base + (size_t)r * DD + kSub;
  }

  v8f acc0 = {}, acc1 = {}, acc2 = {}, acc3 = {};
#pragma unroll 4
  for (int k0 = 0; k0 < DD; k0 += 4) {
    v2f av = *(const v2f*)(aPtr + k0);
    v2f b0 = *(const v2f*)(bPtr[0] + k0);
    v2f b1 = *(const v2f*)(bPtr[1] + k0);
    v2f b2 = *(const v2f*)(bPtr[2] + k0);
    v2f b3 = *(const v2f*)(bPtr[3] + k0);
    acc0 = __builtin_amdgcn_wmma_f32_16x16x4_f32(false, av, false, b0, (short)0, acc0, false, false);
    acc1 = __builtin_amdgcn_wmma_f32_16x16x4_f32(false, av, false, b1, (short)0, acc1, false, false);
    acc2 = __builtin_amdgcn_wmma_f32_16x16x4_f32(false, av, false, b2, (short)0, acc2, false, false);
    acc3 = __builtin_amdgcn_wmma_f32_16x16x4_f32(false, av, false, b3, (short)0, acc3, false, false);
  }

  // D layout: VGPR v -> row (mtile*16 + 8*halfSel + v), col = ntile*16 + sub
  {
    const int rb = mtile * 16 + 8 * halfSel;
    const int c0 = (ng * 4) * 16 + sub;
#pragma unroll
    for (int v = 0; v < 8; ++v) {
      Lg[rb + v][c0 +  0] = acc0[v];
      Lg[rb + v][c0 + 16] = acc1[v];
      Lg[rb + v][c0 + 32] = acc2[v];
      Lg[rb + v][c0 + 48] = acc3[v];
    }
  }

  // ---------------- node logits: 4 partial dots per node -------------------------
  {
    const int node = tid >> 2, part = tid & 3;
    const float* vp = values + (size_t)(g * NPG + node) * DD + part * 64;
    const float* wp = Wnode + part * 64;
    float s = 0.f;
#pragma unroll 8
    for (int j = 0; j < 64; ++j) s = fmaf(vp[j], wp[j], s);
    nlPart[node][part] = s;
  }
  __syncthreads();
  if (tid < NPG)
    nodeLogit[tid] = nlPart[tid][0] + nlPart[tid][1] + nlPart[tid][2] + nlPart[tid][3];
  __syncthreads();

  // ---------------- p_n: 64-wide softmax on wave 0 -------------------------------
  if (w == 0) {
    float x0 = nodeLogit[lane], x1 = nodeLogit[lane + 32];
    float M  = waveRedMax(fmaxf(x0, x1));
    float e0 = expf(x0 - M), e1 = expf(x1 - M);
    float s  = waveRedSum(e0 + e1);
    p_n[lane] = e0 / s;  p_n[lane + 32] = e1 / s;
  }
  __syncthreads();

  // ---------------- row softmax p(a|n) fused with p_a segment-sum ----------------
  {
    float paAcc = 0.f;
#pragma unroll
    for (int i = 0; i < 8; ++i) {
      int n  = w * 8 + i;
      int gn = g * NPG + n;
      float ml = atm[gn * AA + lane] ? Lg[n][lane] : NEGV;
      float M  = waveRedMax(ml);
      float e  = expf(ml - M);
      float s  = waveRedSum(e);
      paAcc = fmaf(p_n[n], e / s, paAcc);
    }
    red[w][lane] = paAcc;
  }
  __syncthreads();
  if (tid < AA) {
    float s = 0.f;
#pragma unroll
    for (int i = 0; i < 8; ++i) s += red[i][tid];
    p_a[tid] = s;
  }
  __syncthreads();

  // ---------------- column segment-softmax p(n|a) + value/entropy terms ----------
  const int ca    = lane;     // action
  const int chunk = w;        // nodes chunk*8 .. chunk*8+7
  float ml8[8], q8[8];
#pragma unroll
  for (int i = 0; i < 8; ++i) {
    int n  = chunk * 8 + i;
    int gn = g * NPG + n;
    int mk = atm[gn * AA + ca] & aam[gn * AA + ca];
    ml8[i] = mk ? Lg[n][32 + ca] : NEGV;
    q8[i]  = 0.5f * (Lg[n][64 + ca] + Lg[n][96 + ca]);
  }
  {
    float mx = ml8[0];
#pragma unroll
    for (int i = 1; i < 8; ++i) mx = fmaxf(mx, ml8[i]);
    red[chunk][ca] = mx;
  }
  __syncthreads();
  if (tid < AA) {
    float m = red[0][tid];
#pragma unroll
    for (int i = 1; i < 8; ++i) m = fmaxf(m, red[i][tid]);
    colmax[tid] = m;
  }
  __syncthreads();
  const float cm = colmax[ca];
  {
    float s = 0.f;
#pragma unroll
    for (int i = 0; i < 8; ++i) s += expf(ml8[i] - cm);
    red[chunk][ca] = s;
  }
  __syncthreads();
  if (tid < AA) {
    float s = 0.f;
#pragma unroll
    for (int i = 0; i < 8; ++i) s += red[i][tid];
    colsum[tid] = s;
  }
  __syncthreads();
  {
    const float cs = colsum[ca];
    float sq = 0.f, hh = 0.f;
#pragma unroll
    for (int i = 0; i < 8; ++i) {
      int n = chunk * 8 + i;
      float p = expf(ml8[i] - cm) / cs;
      out_pna[(size_t)(g * NPG + n) * AA + ca] = p;
      Lg[n][ca] = p;                         // keep for logprob gather (agn cols done)
      sq = fmaf(p, q8[i], sq);
      hh = fmaf(p, logf(p + EPSV), hh);
    }
    red[chunk][ca]  = sq;
    red2[chunk][ca] = hh;
  }
  __syncthreads();
  if (tid < AA) {
    float s = 0.f, h = 0.f;
#pragma unroll
    for (int i = 0; i < 8; ++i) { s += red[i][tid]; h += red2[i][tid]; }
    sqp[tid]  = s;
    hsum[tid] = -h;                          // h_n[g,a]
  }
  __syncthreads();

  // ---------------- scalars: value, entropy, logprob -----------------------------
  if (w == 0) {
    float pa = p_a[lane];
    out_pa[(size_t)g * AA + lane] = pa;
    float valS = waveRedSum(pa * sqp[lane]);
    float entA = -waveRedSum(pa * logf(pa + EPSV));
    float ent2 = waveRedSum(pa * hsum[lane]);
    if (lane == 0) {
      int a_act  = act[2 * g];
      int a_node = act[2 * g + 1] - g * NPG;   // local node index
      out_value[g]   = valS;
      out_entropy[g] = entA + ent2;
      out_logprob[g] = logf(p_a[a_act] + EPSV) + logf(Lg[a_node][a_act] + EPSV);
    }
  }
}

extern "C" void kernel_launch(void* const* d_in, const int* in_sizes, int n_in,
                              void* d_out, int out_size, void* d_ws, size_t ws_size,
                              hipStream_t stream) {
  (void)in_sizes; (void)n_in; (void)d_ws; (void)ws_size; (void)out_size;
  const int*   act    = (const int*)  d_in[0];
  const float* values = (const float*)d_in[1];
  // d_in[2] = indices (implicit i/64), d_in[5] = n_nodes (==64): unused
  const int*   atm    = (const int*)  d_in[3];
  const int*   aam    = (const int*)  d_in[4];
  const float* Wnode  = (const float*)d_in[6];
  const float* Wagn   = (const float*)d_in[7];
  const float* Wnga   = (const float*)d_in[8];
  const float* Wq     = (const float*)d_in[9];

  float* out = (float*)d_out;
  float* out_logprob = out;
  float* out_entropy = out + N_GRAPHS;
  float* out_value   = out + 2 * N_GRAPHS;
  float* out_pa      = out + 3 * N_GRAPHS;
  float* out_pna     = out + 3 * N_GRAPHS + N_GRAPHS * AA;

  atnp_policy_kernel<<<N_GRAPHS, 256, 0, stream>>>(
      act, values, atm, aam, Wnode, Wagn, Wnga, Wq,
      out_logprob, out_entropy, out_value, out_pa, out_pna);
}